// FFTConv_70635032150570
// MI455X (gfx1250) — compile-verified
//
#include <hip/hip_runtime.h>
#include <hip/hip_bf16.h>
#include <math.h>
#include <stdint.h>

// Problem dims
#define BB 16
#define HH 256
#define LL 8192
#define PP 64
#define CH 16                 // chunk (time steps) == WMMA M
#define TCH 16                // chunks per tile == WMMA N
#define TILE (CH*TCH)         // 256 time steps per tile
#define NTILE (LL/TILE)       // 32 tiles per row

typedef __attribute__((ext_vector_type(2))) float v2f;
typedef __attribute__((ext_vector_type(8))) float v8f;
typedef __attribute__((ext_vector_type(4))) unsigned int v4u;
typedef __attribute__((ext_vector_type(8))) int v8i;
typedef __attribute__((ext_vector_type(4))) int v4i;

#if __has_builtin(__builtin_amdgcn_wmma_f32_16x16x4_f32)
static __device__ inline v8f wmma4(v2f a, v2f b, v8f c) {
  // D(16x16,f32) = A(16x4,f32) * B(4x16,f32) + C  -> v_wmma_f32_16x16x4_f32
  return __builtin_amdgcn_wmma_f32_16x16x4_f32(false, a, false, b, (short)0, c,
                                               false, false);
}
#else
static __device__ inline v8f wmma4(v2f a, v2f b, v8f c) {
  for (int e = 0; e < 8; ++e) c[e] += a.x * b.x + a.y * b.y;
  return c;
}
#endif

// ---------------------------------------------------------------------------
// Tensor Data Mover: 1-D tile DMA between global and LDS (TENSORcnt-tracked).
// D# layout per CDNA5 ISA.
// ---------------------------------------------------------------------------
#if __has_builtin(__builtin_amdgcn_tensor_load_to_lds) && \
    __has_builtin(__builtin_amdgcn_s_wait_tensorcnt)
#define HAVE_TDM_LD 1
#endif
#if __has_builtin(__builtin_amdgcn_tensor_store_from_lds) && \
    __has_builtin(__builtin_amdgcn_s_wait_tensorcnt)
#define HAVE_TDM_ST 1
#endif

#if defined(HAVE_TDM_LD) || defined(HAVE_TDM_ST)
static __device__ inline v4u tdm_group0(uint32_t lds_addr, const void* gptr) {
  const uint64_t ga = (uint64_t)(uintptr_t)gptr;
  v4u g0;
  g0[0] = 1u;                                    // count=1, user descriptor
  g0[1] = lds_addr;                              // LDS byte address
  g0[2] = (uint32_t)ga;                          // global_addr[31:0]
  g0[3] = (uint32_t)((ga >> 32) & 0x01FFFFFFu)   // global_addr[56:32]
        | (2u << 30);                            // type=2 ("image")
  return g0;
}
static __device__ inline v8i tdm_group1(uint32_t nelem) {
  v8i g1;
  g1[0] = (int)(2u << 16);                       // wg_mask=0, data_size=4B
  g1[1] = (int)((nelem & 0xFFFFu) << 16);        // tensor_dim0[15:0]
  g1[2] = (int)(((nelem >> 16) & 0xFFFFu)        // tensor_dim0[31:16]
        | (1u << 16));                           // tensor_dim1 = 1
  g1[3] = (int)((nelem & 0xFFFFu) << 16);        // tile_dim0 = nelem
  g1[4] = 1;                                     // tile_dim1 = 1, tile_dim2 = 0
  g1[5] = (int)nelem;                            // tensor_dim0_stride[31:0]
  g1[6] = (int)((nelem & 0xFFFFu) << 16);        // tensor_dim1_stride[15:0]
  g1[7] = (int)(nelem >> 16);                    // tensor_dim1_stride[47:16]
  return g1;
}
#endif

#ifdef HAVE_TDM_LD
static __device__ inline void tdm_load_row(uint32_t lds_addr, const float* gptr,
                                           uint32_t nelem) {
  const v4u g0 = tdm_group0(lds_addr, gptr);
  const v8i g1 = tdm_group1(nelem);
  const v4i z = {0, 0, 0, 0};
#if defined(__clang_major__) && (__clang_major__ >= 23)
  const v8i z8 = {0, 0, 0, 0, 0, 0, 0, 0};
  __builtin_amdgcn_tensor_load_to_lds(g0, g1, z, z, z8, 0);
#else
  __builtin_amdgcn_tensor_load_to_lds(g0, g1, z, z, 0);
#endif
}
#endif

#ifdef HAVE_TDM_ST
static __device__ inline void tdm_store_row(uint32_t lds_addr, float* gptr,
                                            uint32_t nelem) {
  const v4u g0 = tdm_group0(lds_addr, gptr);
  const v8i g1 = tdm_group1(nelem);
  const v4i z = {0, 0, 0, 0};
#if defined(__clang_major__) && (__clang_major__ >= 23)
  const v8i z8 = {0, 0, 0, 0, 0, 0, 0, 0};
  __builtin_amdgcn_tensor_store_from_lds(g0, g1, z, z, z8, 0);
#else
  __builtin_amdgcn_tensor_store_from_lds(g0, g1, z, z, 0);
#endif
}
#endif

// ---------------------------------------------------------------------------
// One-time table build:
//   Tkern[h][16][16] : lower-tri Toeplitz of kshort[d] = Re(sum_p BC_p A_p^d)
//   Mcar [h][16][128]: row i, col 2p -> Re(BC_p A_p^{i+1}), 2p+1 -> -Im(...)
//   Wt   [128][16]   : row 2p -> Re(A_p^{15-j}), 2p+1 -> Im(A_p^{15-j})
//   A16  [128]       : (Re,Im) of A_p^16
// ---------------------------------------------------------------------------
__global__ void fftconv_tables(const float* __restrict__ A_re,
                               const float* __restrict__ A_im,
                               const float* __restrict__ BC_re,
                               const float* __restrict__ BC_im,
                               float* __restrict__ Tkern,
                               float* __restrict__ Mcar,
                               float* __restrict__ Wt,
                               float* __restrict__ A16) {
  const int h = blockIdx.x;
  const int tid = threadIdx.x; // 128 threads

  for (int idx = tid; idx < 16 * 128; idx += blockDim.x) {
    const int i = idx >> 7;
    const int q = idx & 127;
    const int p = q >> 1;
    const float ar = A_re[p], ai = A_im[p];
    float pr = 1.f, pi = 0.f;
    for (int e = 0; e <= i; ++e) { const float nr = pr*ar - pi*ai; pi = pr*ai + pi*ar; pr = nr; }
    const float bcr = BC_re[h*PP + p], bci = BC_im[h*PP + p];
    const float vr = bcr*pr - bci*pi;   // Re(BC * A^{i+1})
    const float vi = bcr*pi + bci*pr;   // Im(BC * A^{i+1})
    Mcar[h*2048 + i*128 + q] = (q & 1) ? -vi : vr;
  }

  for (int idx = tid; idx < 256; idx += blockDim.x) {
    const int i = idx >> 4, j = idx & 15;
    float v = 0.f;
    if (j <= i) {
      const int d = i - j;
      for (int p = 0; p < PP; ++p) {
        const float ar = A_re[p], ai = A_im[p];
        float pr = 1.f, pi = 0.f;
        for (int e = 0; e < d; ++e) { const float nr = pr*ar - pi*ai; pi = pr*ai + pi*ar; pr = nr; }
        v += BC_re[h*PP + p]*pr - BC_im[h*PP + p]*pi;
      }
    }
    Tkern[h*256 + idx] = v;
  }

  if (h == 0) {
    for (int idx = tid; idx < 128 * 16; idx += blockDim.x) {
      const int q = idx >> 4, j = idx & 15;
      const int p = q >> 1, e = 15 - j;
      const float ar = A_re[p], ai = A_im[p];
      float pr = 1.f, pi = 0.f;
      for (int k = 0; k < e; ++k) { const float nr = pr*ar - pi*ai; pi = pr*ai + pi*ar; pr = nr; }
      Wt[q*16 + j] = (q & 1) ? pi : pr;
    }
    if (tid < PP) {
      const float ar = A_re[tid], ai = A_im[tid];
      float pr = 1.f, pi = 0.f;
      for (int k = 0; k < 16; ++k) { const float nr = pr*ar - pi*ai; pi = pr*ai + pi*ar; pr = nr; }
      A16[2*tid] = pr; A16[2*tid + 1] = pi;
    }
  }
}

// ---------------------------------------------------------------------------
// Main: one block (64 threads = 2 waves, wave32) per (b,h) row.
// Software-pipelined TDM: tile t+1 DMA overlaps tile t WMMA work.
// TENSORcnt retires in issue order (L0,L1,S0,L2,S1,...), so:
//   after issuing L_{t+1}:  wait tensorcnt<=2  => L_t complete
//   before restaging Yout:  wait tensorcnt<=1  => S_{t-1} complete
// ---------------------------------------------------------------------------
__global__ __launch_bounds__(64) void fftconv_main(
    const float* __restrict__ u, const float* __restrict__ Dvec,
    const float* __restrict__ Tkern, const float* __restrict__ Mcar,
    const float* __restrict__ Wt, const float* __restrict__ A16,
    float* __restrict__ out) {
  __shared__ float lds_ub[2][TILE];  // 2 KB : double-buffered input tiles
  __shared__ float lds_Z[128*TCH];   // 8 KB : chunk state increments
  __shared__ float lds_C[128*TCH];   // 8 KB : carry matrix (state entering chunk)
  __shared__ float lds_Yi[256];      // 1 KB : intra-chunk result
  __shared__ float lds_Yo[256];      // 1 KB : output tile staging (TDM source)

  const int bh   = blockIdx.x;
  const int h    = bh & (HH - 1);
  const int tid  = threadIdx.x;
  const int wave = tid >> 5;
  const int lane = tid & 31;
  const int lo   = lane & 15;        // WMMA row/col within half-wave
  const int hi2  = (lane >> 4) << 1; // K offset for upper half-wave
  const int mb   = (lane >> 4) << 3; // D-matrix M base per half-wave

  const float* __restrict__ urow = u + (size_t)bh * LL;
  float* __restrict__ orow = out + (size_t)bh * LL;

  const float Dh   = Dvec[h];
  const float a16r = A16[2*tid];
  const float a16i = A16[2*tid + 1];

  __builtin_prefetch(&Mcar[h*2048], 0, 1);   // global_prefetch_b8
  __builtin_prefetch(&Tkern[h*256], 0, 1);

  // ---- prologue: start DMA of tile 0 -------------------------------------
#ifdef HAVE_TDM_LD
  if (wave == 0) {
    tdm_load_row((uint32_t)(uintptr_t)(void*)lds_ub[0], urow, TILE);
  }
#else
  {
    const float4* src = reinterpret_cast<const float4*>(urow);
    reinterpret_cast<float4*>(lds_ub[0])[tid] = src[tid];
  }
#endif

  float carry_re = 0.f, carry_im = 0.f;   // |A^L| ~ 1e-36 -> circular wrap ~ 0

  for (int t = 0; t < NTILE; ++t) {
    const float* __restrict__ ubuf = lds_ub[t & 1];

    // ---- prefetch tile t+1, ensure tile t landed (in-order TENSORcnt) ----
#ifdef HAVE_TDM_LD
    if (wave == 0) {
      if (t + 1 < NTILE)
        tdm_load_row((uint32_t)(uintptr_t)(void*)lds_ub[(t + 1) & 1],
                     urow + (t + 1) * TILE, TILE);
      if (t == 0 || t + 1 == NTILE)
        __builtin_amdgcn_s_wait_tensorcnt(1);  // only newest op may remain
      else
        __builtin_amdgcn_s_wait_tensorcnt(2);  // L_{t+1}, S_{t-1} may remain
    }
#else
    if (t + 1 < NTILE) {
      const float4* src =
          reinterpret_cast<const float4*>(urow + (t + 1) * TILE);
      reinterpret_cast<float4*>(lds_ub[(t + 1) & 1])[tid] = src[tid];
    }
#endif
    __syncthreads();

    // ---- Z(128x16) = Wt(128x16) @ U(16x16): 4 slabs per wave -------------
    for (int s = wave*4; s < wave*4 + 4; ++s) {
      const int r0 = s * 16;
      v8f acc = {};
      for (int kk = 0; kk < 4; ++kk) {
        const int kb = kk*4 + hi2;
        v2f a, b;
        a.x = Wt[(r0 + lo)*16 + kb];
        a.y = Wt[(r0 + lo)*16 + kb + 1];
        b.x = ubuf[lo*16 + kb];
        b.y = ubuf[lo*16 + kb + 1];
        acc = wmma4(a, b, acc);
      }
      for (int e = 0; e < 8; ++e)
        lds_Z[(r0 + mb + e)*16 + lo] = acc[e];
    }
    __syncthreads();

    // ---- sequential diagonal carry scan over 16 chunks (1 state/thread) --
    {
      const int p2 = tid * 2;
      float cr = carry_re, ci = carry_im;
      for (int c = 0; c < TCH; ++c) {
        lds_C[p2*16 + c]     = cr;
        lds_C[(p2+1)*16 + c] = ci;
        const float zr = lds_Z[p2*16 + c];
        const float zi = lds_Z[(p2+1)*16 + c];
        const float nr = a16r*cr - a16i*ci + zr;
        const float ni = a16r*ci + a16i*cr + zi;
        cr = nr; ci = ni;
      }
      carry_re = cr; carry_im = ci;
    }
    __syncthreads();

    v8f yacc = {};
    if (wave == 1) {
      // ---- Yintra(16x16) = Tkern_h(16x16) @ U(16x16) ----------------------
      v8f acc = {};
      for (int kk = 0; kk < 4; ++kk) {
        const int kb = kk*4 + hi2;
        v2f a, b;
        a.x = Tkern[h*256 + lo*16 + kb];
        a.y = Tkern[h*256 + lo*16 + kb + 1];
        b.x = ubuf[lo*16 + kb];
        b.y = ubuf[lo*16 + kb + 1];
        acc = wmma4(a, b, acc);
      }
      for (int e = 0; e < 8; ++e)
        lds_Yi[(mb + e)*16 + lo] = acc[e];
    } else {
      // ---- Ycar(16x16) = Mcar_h(16x128) @ Carry(128x16) -------------------
      for (int kk = 0; kk < 32; ++kk) {
        const int kb = kk*4 + hi2;
        v2f a, b;
        a.x = Mcar[h*2048 + lo*128 + kb];
        a.y = Mcar[h*2048 + lo*128 + kb + 1];
        b.x = lds_C[kb*16 + lo];
        b.y = lds_C[(kb+1)*16 + lo];
        yacc = wmma4(a, b, yacc);
      }
    }
    __syncthreads();

    // ---- combine + D*u skip + tanh ---------------------------------------
    if (wave == 0) {
#ifdef HAVE_TDM_ST
      if (t + 1 < NTILE)
        __builtin_amdgcn_s_wait_tensorcnt(1);  // S_{t-1} done (L_{t+1} may fly)
      else
        __builtin_amdgcn_s_wait_tensorcnt(0);  // nothing issued after S_{t-1}
#endif
      float tmp[8];
      for (int e = 0; e < 8; ++e) {
        const int m = mb + e;
        const float uval = ubuf[lo*16 + m];
        const float y = yacc[e] + lds_Yi[m*16 + lo] + Dh * uval;
        tmp[e] = tanhf(y);
      }
#ifdef HAVE_TDM_ST
      for (int e = 0; e < 8; ++e)
        lds_Yo[lo*16 + mb + e] = tmp[e];
#else
      float* op = orow + t * TILE + lo*16 + mb;
      reinterpret_cast<float4*>(op)[0] = make_float4(tmp[0], tmp[1], tmp[2], tmp[3]);
      reinterpret_cast<float4*>(op)[1] = make_float4(tmp[4], tmp[5], tmp[6], tmp[7]);
#endif
    }
    __syncthreads();   // barrier waits dscnt: lds_Yo writes visible to TDM

#ifdef HAVE_TDM_ST
    if (wave == 0)
      tdm_store_row((uint32_t)(uintptr_t)(void*)lds_Yo, orow + t * TILE, TILE);
#endif
  }

#if defined(HAVE_TDM_LD) || defined(HAVE_TDM_ST)
  if (wave == 0) __builtin_amdgcn_s_wait_tensorcnt(0);   // drain before end
#endif
}

extern "C" void kernel_launch(void* const* d_in, const int* in_sizes, int n_in,
                              void* d_out, int out_size, void* d_ws, size_t ws_size,
                              hipStream_t stream) {
  const float* u     = (const float*)d_in[0];
  const float* A_re  = (const float*)d_in[1];
  const float* A_im  = (const float*)d_in[2];
  const float* BC_re = (const float*)d_in[3];
  const float* BC_im = (const float*)d_in[4];
  const float* Dv    = (const float*)d_in[5];
  float* out = (float*)d_out;

  float* ws    = (float*)d_ws;
  float* Tkern = ws;                    // 256*256   = 65536 f32
  float* Mcar  = Tkern + 256*256;       // 256*2048  = 524288 f32
  float* Wt    = Mcar + 256*2048;       // 2048 f32
  float* A16   = Wt + 2048;             // 128 f32

  fftconv_tables<<<HH, 128, 0, stream>>>(A_re, A_im, BC_re, BC_im,
                                         Tkern, Mcar, Wt, A16);
  fftconv_main<<<BB*HH, 64, 0, stream>>>(u, Dv, Tkern, Mcar, Wt, A16, out);
}